// Net_26774826123689
// MI455X (gfx1250) — compile-verified
//
#include <hip/hip_runtime.h>
#include <math.h>

// ---------------------------------------------------------------------------
// SplineConv GNN for MI455X (gfx1250, wave32).
// Layer 2: scatter Z_k[dst] += basis*h1[src] (f32 atomics), then
// agg2 = sum_k Z_k @ W2_k + h1 @ root2 via v_wmma_f32_16x16x4_f32.
// W2/root2 staged in LDS *pair-interleaved* so each WMMA B operand is one
// aligned ds_load_b64 (no VGPR repack); staging uses
// global_load_async_to_lds_b32 (ASYNCcnt path) — builtin confirmed present.
// ---------------------------------------------------------------------------

typedef float v2f __attribute__((ext_vector_type(2)));
typedef float v8f __attribute__((ext_vector_type(8)));

#if defined(__has_builtin)
#if __has_builtin(__builtin_amdgcn_global_load_async_to_lds_b32)
#define USE_ASYNC_LDS 1
typedef __attribute__((address_space(1))) int gas_int;   // global addrspace
typedef __attribute__((address_space(3))) int las_int;   // LDS addrspace
#endif
#endif

__device__ __forceinline__ float elu1(float v) { return v > 0.0f ? v : expm1f(v); }

// ---- zero init (graph-capture safe memset) --------------------------------
__global__ __launch_bounds__(256) void k_zero4(float4* p, long n4) {
  long i = (long)blockIdx.x * blockDim.x + threadIdx.x;
  long stride = (long)gridDim.x * blockDim.x;
  float4 z; z.x = z.y = z.z = z.w = 0.0f;
  for (; i < n4; i += stride) p[i] = z;
}

// ---- spline basis + kernel indices + degree -------------------------------
__global__ __launch_bounds__(256) void k_prep(const float* __restrict__ ea,
                                              const int* __restrict__ ei,
                                              float* __restrict__ basis,
                                              int* __restrict__ idx4,
                                              float* __restrict__ deg, int E) {
  int e = blockIdx.x * blockDim.x + threadIdx.x;
  if (e >= E) return;
  float v0 = ea[2 * e] * 4.0f, v1 = ea[2 * e + 1] * 4.0f;   // (K-1)=4
  float b0 = fminf(fmaxf(floorf(v0), 0.0f), 3.0f);
  float b1 = fminf(fmaxf(floorf(v1), 0.0f), 3.0f);
  float f0 = v0 - b0, f1 = v1 - b1;
  int i00 = (int)b0 + 5 * (int)b1;
  float4 bs;
  bs.x = (1.0f - f0) * (1.0f - f1);
  bs.y = f0 * (1.0f - f1);
  bs.z = (1.0f - f0) * f1;
  bs.w = f0 * f1;
  int4 id; id.x = i00; id.y = i00 + 1; id.z = i00 + 5; id.w = i00 + 6;
  ((float4*)basis)[e] = bs;
  ((int4*)idx4)[e] = id;
  atomicAdd(&deg[ei[E + e]], 1.0f);
}

__global__ __launch_bounds__(256) void k_rdeg(const float* __restrict__ deg,
                                              float* __restrict__ rdeg, int N) {
  int n = blockIdx.x * blockDim.x + threadIdx.x;
  if (n < N) rdeg[n] = 1.0f / fmaxf(deg[n], 1.0f);
}

// ---- layer 1: one wave per edge, lane = output channel (wave32 fit) -------
__global__ __launch_bounds__(256) void k_l1_edge(const float* __restrict__ x,
                                                 const int* __restrict__ ei,
                                                 const float* __restrict__ basis,
                                                 const int* __restrict__ idx4,
                                                 const float* __restrict__ W1,
                                                 float* __restrict__ agg1, int E) {
  int gt = blockIdx.x * blockDim.x + threadIdx.x;
  int e = gt >> 5, o = gt & 31;
  if (e >= E) return;
  int src = ei[e], dst = ei[E + e];
  float x0 = x[src * 3], x1 = x[src * 3 + 1], x2 = x[src * 3 + 2];
  float4 bs = ((const float4*)basis)[e];
  int4 id = ((const int4*)idx4)[e];
  float msg = 0.0f;
  { const float* Wk = W1 + id.x * 96; msg += bs.x * (x0 * Wk[o] + x1 * Wk[32 + o] + x2 * Wk[64 + o]); }
  { const float* Wk = W1 + id.y * 96; msg += bs.y * (x0 * Wk[o] + x1 * Wk[32 + o] + x2 * Wk[64 + o]); }
  { const float* Wk = W1 + id.z * 96; msg += bs.z * (x0 * Wk[o] + x1 * Wk[32 + o] + x2 * Wk[64 + o]); }
  { const float* Wk = W1 + id.w * 96; msg += bs.w * (x0 * Wk[o] + x1 * Wk[32 + o] + x2 * Wk[64 + o]); }
  atomicAdd(&agg1[dst * 32 + o], msg);
}

__global__ __launch_bounds__(256) void k_l1_node(const float* __restrict__ x,
                                                 const float* __restrict__ agg1,
                                                 const float* __restrict__ rdeg,
                                                 const float* __restrict__ root1,
                                                 const float* __restrict__ b1,
                                                 float* __restrict__ h1, int N) {
  int t = blockIdx.x * blockDim.x + threadIdx.x;
  int n = t >> 5, o = t & 31;
  if (n >= N) return;
  float v = agg1[n * 32 + o] * rdeg[n]
          + x[n * 3] * root1[o] + x[n * 3 + 1] * root1[32 + o] + x[n * 3 + 2] * root1[64 + o]
          + b1[o];
  h1[n * 32 + o] = elu1(v);
}

// ---- layer 2 scatter: Z[dst, idx_s, o] += basis_s * h1[src, o] ------------
__global__ __launch_bounds__(256) void k_l2_scatter(const float* __restrict__ h1,
                                                    const int* __restrict__ ei,
                                                    const float* __restrict__ basis,
                                                    const int* __restrict__ idx4,
                                                    float* __restrict__ Z, int E) {
  int gt = blockIdx.x * blockDim.x + threadIdx.x;
  int e = gt >> 5, o = gt & 31;
  if (e >= E) return;
  int src = ei[e], dst = ei[E + e];
  float v = h1[src * 32 + o];
  float4 bs = ((const float4*)basis)[e];
  int4 id = ((const int4*)idx4)[e];
  float* zb = Z + (size_t)dst * 800 + o;
  atomicAdd(zb + id.x * 32, bs.x * v);
  atomicAdd(zb + id.y * 32, bs.y * v);
  atomicAdd(zb + id.z * 32, bs.z * v);
  atomicAdd(zb + id.w * 32, bs.w * v);
}

// ---- LDS staging helper: pair-interleave K rows ---------------------------
// Source (global): g[(i/1024)*2048 + ((i%1024)/64)*128 + (i%64)] and +64.
// Dest (LDS, as float2): sWp[i] = { K=2kb row, K=2kb+1 row } at column col.
__device__ __forceinline__ void stage_pairs(const float* __restrict__ g,
                                            float* __restrict__ sWf,
                                            int npairs, int tid) {
  for (int i = tid; i < npairs; i += 256) {
    int base = (i >> 10) * 2048 + (((i >> 6) & 15) << 7) + (i & 63);
#if defined(USE_ASYNC_LDS)
    __builtin_amdgcn_global_load_async_to_lds_b32(
        (gas_int*)(float*)(g + base),
        (las_int*)(sWf + 2 * i), 0, 0);
    __builtin_amdgcn_global_load_async_to_lds_b32(
        (gas_int*)(float*)(g + base + 64),
        (las_int*)(sWf + 2 * i + 1), 0, 0);
#else
    v2f p;
    p.x = g[base];
    p.y = g[base + 64];
    ((v2f*)sWf)[i] = p;
#endif
  }
#if defined(USE_ASYNC_LDS)
  asm volatile("s_wait_asynccnt 0" ::: "memory");
#endif
}

// ---- layer 2 GEMM via WMMA fp32: agg2 = sum_k Z_k@W2_k ; += h1@root2 ------
// One wave per 16-node M-tile; 4 persistent 16x16 f32 accumulators (N=64).
// W2 staged pair-interleaved into LDS in 5-kernel (40KB) chunks, root2 last;
// each B operand is a single aligned ds_load_b64.
__global__ __launch_bounds__(256) void k_l2_gemm(const float* __restrict__ Z,
                                                 const float* __restrict__ h1,
                                                 const float* __restrict__ W2,
                                                 const float* __restrict__ root2,
                                                 const float* __restrict__ b2,
                                                 const float* __restrict__ rdeg,
                                                 float* __restrict__ h2,
                                                 int N, int Mtiles) {
  __shared__ v2f sWp[5 * 16 * 64];            // 40 KB: [k][kb=K/2][col] pairs
  float* sWf = (float*)sWp;
  int wave = threadIdx.x >> 5, lane = threadIdx.x & 31;
  int m = blockIdx.x * 8 + wave;
  bool valid = m < Mtiles;
  int n0 = m * 16;
  int row = lane & 15, half = lane >> 4;      // A/B: lane -> (M or Ncol)=row, K offset 2*half

  v8f C[4];
  #pragma unroll
  for (int t = 0; t < 4; t++)
    #pragma unroll
    for (int r = 0; r < 8; r++) C[t][r] = 0.0f;

  for (int p = 0; p < 5; p++) {               // 5 chunks x 5 kernels = 25
    __syncthreads();
    stage_pairs(W2 + p * 10240, sWf, 5120, threadIdx.x);
    __syncthreads();
    if (valid) {
      for (int kl = 0; kl < 5; kl++) {
        v2f A[8];
        const float* zr = Z + (size_t)(n0 + row) * 800 + (p * 5 + kl) * 32 + 2 * half;
        #pragma unroll
        for (int c = 0; c < 8; c++) A[c] = *(const v2f*)(zr + 4 * c);
        const v2f* bw = sWp + (kl * 16 + half) * 64;   // pair kb = 2c + half
        #pragma unroll
        for (int t = 0; t < 4; t++) {
          int col = t * 16 + row;
          #pragma unroll
          for (int c = 0; c < 8; c++) {
            v2f B = bw[(2 * c) * 64 + col];
            C[t] = __builtin_amdgcn_wmma_f32_16x16x4_f32(
                false, A[c], false, B, (short)0, C[t], false, false);
          }
        }
      }
    }
  }

  // mean-normalize (divide by degree) BEFORE adding the root term
  if (valid) {
    float rdv[8];
    #pragma unroll
    for (int r = 0; r < 8; r++) {
      int n = n0 + 8 * half + r;              // C element M = r + 8*half
      rdv[r] = (n < N) ? rdeg[n] : 0.0f;
    }
    #pragma unroll
    for (int t = 0; t < 4; t++)
      #pragma unroll
      for (int r = 0; r < 8; r++) C[t][r] *= rdv[r];
  }

  // root term: C += h1_tile @ root2 (8 more WMMAs per t-tile)
  __syncthreads();
  stage_pairs(root2, sWf, 1024, threadIdx.x);
  __syncthreads();
  if (valid) {
    v2f A[8];
    const float* hr = h1 + (size_t)(n0 + row) * 32 + 2 * half;
    #pragma unroll
    for (int c = 0; c < 8; c++) A[c] = *(const v2f*)(hr + 4 * c);
    const v2f* bw = sWp + half * 64;
    #pragma unroll
    for (int t = 0; t < 4; t++) {
      int col = t * 16 + row;
      #pragma unroll
      for (int c = 0; c < 8; c++) {
        v2f B = bw[(2 * c) * 64 + col];
        C[t] = __builtin_amdgcn_wmma_f32_16x16x4_f32(
            false, A[c], false, B, (short)0, C[t], false, false);
      }
    }
    // bias + ELU + store h2
    #pragma unroll
    for (int t = 0; t < 4; t++) {
      float bb = b2[t * 16 + row];
      #pragma unroll
      for (int r = 0; r < 8; r++) {
        int n = n0 + 8 * half + r;
        if (n < N) h2[(size_t)n * 64 + t * 16 + row] = elu1(C[t][r] + bb);
      }
    }
  }
}

// ---- batch mean-pool: block partials in LDS (ds_add_f32), then global -----
__global__ __launch_bounds__(256) void k_pool(const float* __restrict__ h2,
                                              const int* __restrict__ batch,
                                              float* __restrict__ pooled, // [512]+[8]
                                              int N) {
  __shared__ float part[520];
  for (int i = threadIdx.x; i < 520; i += 256) part[i] = 0.0f;
  __syncthreads();
  int col = threadIdx.x & 63, r0 = threadIdx.x >> 6;
  int n0 = blockIdx.x * 128;
  for (int nl = r0; nl < 128; nl += 4) {
    int n = n0 + nl;
    if (n < N) {
      int b = batch[n];
      atomicAdd(&part[b * 64 + col], h2[(size_t)n * 64 + col]);
      if (col == 0) atomicAdd(&part[512 + b], 1.0f);
    }
  }
  __syncthreads();
  for (int i = threadIdx.x; i < 520; i += 256) atomicAdd(&pooled[i], part[i]);
}

// ---- head MLP + log_softmax (1-wide axis => exactly 0, computed anyway) ---
__global__ __launch_bounds__(128) void k_head(const float* __restrict__ pooledcnt,
                                              const float* __restrict__ lw1,
                                              const float* __restrict__ lb1,
                                              const float* __restrict__ lw2,
                                              const float* __restrict__ lb2,
                                              float* __restrict__ out) {
  __shared__ float pm[512];
  __shared__ float hid[1024];
  for (int i = threadIdx.x; i < 512; i += 128) {
    int b = i >> 6;
    pm[i] = pooledcnt[i] / fmaxf(pooledcnt[512 + b], 1.0f);
  }
  __syncthreads();
  int j = threadIdx.x;
  for (int b = 0; b < 8; b++) {
    float s = lb1[j];
    for (int c = 0; c < 64; c++) s += pm[b * 64 + c] * lw1[c * 128 + j];
    hid[b * 128 + j] = elu1(s);
  }
  __syncthreads();
  if (threadIdx.x < 8) {
    int b = threadIdx.x;
    float lg = lb2[0];
    for (int jj = 0; jj < 128; jj++) lg += hid[b * 128 + jj] * lw2[jj];
    float mx = lg;
    float lse = mx + logf(expf(lg - mx));   // single class
    out[b] = lg - lse;
  }
}

// ---------------------------------------------------------------------------
extern "C" void kernel_launch(void* const* d_in, const int* in_sizes, int n_in,
                              void* d_out, int out_size, void* d_ws, size_t ws_size,
                              hipStream_t stream) {
  (void)n_in; (void)out_size; (void)ws_size;
  const float* x     = (const float*)d_in[0];
  const int*   ei    = (const int*)d_in[1];
  const float* ea    = (const float*)d_in[2];
  const int*   batch = (const int*)d_in[3];
  const float* W1    = (const float*)d_in[4];
  const float* root1 = (const float*)d_in[5];
  const float* b1    = (const float*)d_in[6];
  const float* W2    = (const float*)d_in[7];
  const float* root2 = (const float*)d_in[8];
  const float* b2    = (const float*)d_in[9];
  const float* lw1   = (const float*)d_in[10];
  const float* lb1   = (const float*)d_in[11];
  const float* lw2   = (const float*)d_in[12];
  const float* lb2   = (const float*)d_in[13];

  const int N = in_sizes[0] / 3;
  const int E = in_sizes[1] / 2;

  float* ws = (float*)d_ws;
  // zero-initialized region (packed contiguously): Z | agg1 | deg | pooled+cnt
  size_t offZ    = 0;
  size_t offA1   = offZ + (size_t)N * 800;
  size_t offDeg  = offA1 + (size_t)N * 32;
  size_t offPool = offDeg + (size_t)N;           // 512 + 8
  size_t zf      = (offPool + 520 + 3) & ~(size_t)3;
  // non-zeroed scratch
  size_t offB  = zf;                              // basis  E*4
  size_t offI  = offB + (size_t)E * 4;            // idx4   E*4 (int)
  size_t offR  = offI + (size_t)E * 4;            // rdeg   N
  size_t offH1 = (offR + (size_t)N + 3) & ~(size_t)3;  // h1 N*32
  size_t offH2 = offH1 + (size_t)N * 32;               // h2 N*64

  k_zero4<<<4096, 256, 0, stream>>>((float4*)ws, (long)(zf / 4));
  k_prep<<<(E + 255) / 256, 256, 0, stream>>>(ea, ei, ws + offB, (int*)(ws + offI),
                                              ws + offDeg, E);
  k_rdeg<<<(N + 255) / 256, 256, 0, stream>>>(ws + offDeg, ws + offR, N);
  k_l1_edge<<<(E * 32 + 255) / 256, 256, 0, stream>>>(x, ei, ws + offB,
                                                      (int*)(ws + offI), W1,
                                                      ws + offA1, E);
  k_l1_node<<<(N * 32 + 255) / 256, 256, 0, stream>>>(x, ws + offA1, ws + offR,
                                                      root1, b1, ws + offH1, N);
  k_l2_scatter<<<(E * 32 + 255) / 256, 256, 0, stream>>>(ws + offH1, ei, ws + offB,
                                                         (int*)(ws + offI),
                                                         ws + offZ, E);
  int Mt = (N + 15) / 16;
  k_l2_gemm<<<(Mt + 7) / 8, 256, 0, stream>>>(ws + offZ, ws + offH1, W2, root2, b2,
                                              ws + offR, ws + offH2, N, Mt);
  k_pool<<<(N + 127) / 128, 256, 0, stream>>>(ws + offH2, batch, ws + offPool, N);
  k_head<<<1, 128, 0, stream>>>(ws + offPool, lw1, lb1, lw2, lb2, (float*)d_out);
}